// TextKernelLoss_13400297963741
// MI455X (gfx1250) — compile-verified
//
#include <hip/hip_runtime.h>
#include <hip/hip_bf16.h>

#define N_BATCH 16
#define HW      409600      // 640*640
#define BPB     64          // blocks per batch for streaming kernels
#define CHUNK   (HW / BPB)  // 6400 elements per block
#define NTHR    256
#define SMOOTHF 1e-6f

typedef float v2f __attribute__((ext_vector_type(2)));
typedef float v8f __attribute__((ext_vector_type(8)));

// monotonic map: larger float  <->  larger uint key (total order, -0 < +0)
__device__ __forceinline__ unsigned keymap(float x) {
    unsigned s = __float_as_uint(x);
    return (s & 0x80000000u) ? ~s : (s | 0x80000000u);
}

__device__ __forceinline__ float sigmoidf_stable(float x) {
    if (x >= 0.0f) { return 1.0f / (1.0f + expf(-x)); }
    float e = expf(x);
    return e / (1.0f + e);
}

// ---------------- zero workspace (hist + counters) ----------------
__global__ void tkl_zero_ws(unsigned* __restrict__ ws, int nwords) {
    int i = blockIdx.x * blockDim.x + threadIdx.x;
    if (i < nwords) ws[i] = 0u;
}

// ---------------- pass 0: counts + top-8-bit histogram ----------------
__global__ void tkl_hist0(const float* __restrict__ preds,
                          const float* __restrict__ targets,
                          const float* __restrict__ eff,
                          unsigned* __restrict__ hist,
                          unsigned* __restrict__ posn,
                          unsigned* __restrict__ negav) {
    __shared__ unsigned lh[256];
    __shared__ unsigned lp, ln;
    const int n = blockIdx.y, tid = threadIdx.x;
    lh[tid] = 0u;
    if (tid == 0) { lp = 0u; ln = 0u; }
    __syncthreads();

    const float* xb = preds   + (size_t)n * 6 * HW;   // channel 0 (text logits)
    const float* tb = targets + (size_t)n * 2 * HW;   // channel 0 (text target)
    const float* eb = eff     + (size_t)n * HW;
    const int base = blockIdx.x * CHUNK;

    unsigned cp = 0u, cn = 0u;
    for (int i = tid; i < CHUNK; i += NTHR) {
        const int hw = base + i;
        const float t = tb[hw];
        if (t > 0.5f) {
            if (eb[hw] > 0.5f) cp++;
        } else {
            cn++;
            atomicAdd(&lh[keymap(xb[hw]) >> 24], 1u);
        }
    }
    atomicAdd(&lp, cp);
    atomicAdd(&ln, cn);
    __syncthreads();

    if (lh[tid]) atomicAdd(&hist[n * 256 + tid], lh[tid]);
    if (tid == 0) { atomicAdd(&posn[n], lp); atomicAdd(&negav[n], ln); }
}

// ---------------- passes 1..3: refine histogram under current prefix ----------------
__global__ void tkl_hist(const float* __restrict__ preds,
                         const float* __restrict__ targets,
                         const unsigned* __restrict__ prefix,
                         unsigned* __restrict__ hist,
                         int shift, unsigned mmask) {
    __shared__ unsigned lh[256];
    const int n = blockIdx.y, tid = threadIdx.x;
    lh[tid] = 0u;
    __syncthreads();

    const unsigned pfx = prefix[n];
    const float* xb = preds   + (size_t)n * 6 * HW;
    const float* tb = targets + (size_t)n * 2 * HW;
    const int base = blockIdx.x * CHUNK;

    for (int i = tid; i < CHUNK; i += NTHR) {
        const int hw = base + i;
        if (tb[hw] <= 0.5f) {
            const unsigned key = keymap(xb[hw]);
            if ((key & mmask) == pfx) atomicAdd(&lh[(key >> shift) & 0xFFu], 1u);
        }
    }
    __syncthreads();
    if (lh[tid]) atomicAdd(&hist[n * 256 + tid], lh[tid]);
}

// ---------------- digit selection (one block per batch) ----------------
__global__ void tkl_select(unsigned* __restrict__ hist,
                           const unsigned* __restrict__ posn,
                           const unsigned* __restrict__ negav,
                           unsigned* __restrict__ prefix,
                           unsigned* __restrict__ krem,
                           unsigned* __restrict__ deg,
                           float* __restrict__ thr,
                           int pass) {
    __shared__ unsigned h[256];
    const int n = blockIdx.x, tid = threadIdx.x;
    h[tid] = hist[n * 256 + tid];
    __syncthreads();

    if (tid == 0) {
        unsigned k, pfx;
        if (pass == 0) {
            const unsigned p = posn[n], na = negav[n];
            unsigned kk = p * 3u;
            if (kk > na) kk = na;
            deg[n] = (p == 0u || kk == 0u) ? 1u : 0u;
            k = kk;
            pfx = 0u;
        } else {
            k = krem[n];
            pfx = prefix[n];
        }
        unsigned digit = 0u;
        if (k > 0u) {
            for (int d = 255; d >= 0; --d) {   // largest digit first (descending rank)
                const unsigned c = h[d];
                if (k <= c) { digit = (unsigned)d; break; }
                k -= c;
            }
        }
        pfx |= digit << (24 - 8 * pass);
        prefix[n] = pfx;
        krem[n]   = k;
        if (pass == 3) {
            const unsigned key = pfx;
            const unsigned s = (key & 0x80000000u) ? (key & 0x7FFFFFFFu) : ~key;
            thr[n] = __uint_as_float(s);       // logit-space threshold
        }
    }
    __syncthreads();
    hist[n * 256 + tid] = 0u;                  // ready for next pass
}

// ---------------- dice partial sums (deterministic, per-block slots) ----------------
__global__ void tkl_dice(const float* __restrict__ preds,
                         const float* __restrict__ targets,
                         const float* __restrict__ eff,
                         const float* __restrict__ thr,
                         const unsigned* __restrict__ deg,
                         float* __restrict__ partials) {
    __shared__ float red[6][NTHR];
    const int n = blockIdx.y, tid = threadIdx.x;
    const float* xb = preds   + (size_t)n * 6 * HW;
    const float* kb = xb + HW;
    const float* tb = targets + (size_t)n * 2 * HW;
    const float* gb = tb + HW;
    const float* eb = eff     + (size_t)n * HW;
    const float tlogit = thr[n];
    const bool  dg = (deg[n] != 0u);
    const int base = blockIdx.x * CHUNK;

    float a0 = 0.f, a1 = 0.f, a2 = 0.f, a3 = 0.f, a4 = 0.f, a5 = 0.f;
    for (int i = tid; i < CHUNK; i += NTHR) {
        const int hw = base + i;
        const float xl = xb[hw], kl = kb[hw];
        const float tt = tb[hw], tk = gb[hw];
        const float e  = eb[hw];
        const float pt = sigmoidf_stable(xl);
        const float pk = sigmoidf_stable(kl);
        const bool pos  = tt > 0.5f;
        const bool effb = e  > 0.5f;
        const float sel = dg ? ((e != 0.0f) ? 1.f : 0.f)
                             : ((((xl >= tlogit) || pos) && effb) ? 1.f : 0.f);
        const float p = pt * sel;
        const float g = (pos ? 1.f : 0.f) * sel;
        a0 += p * g;  a1 += p * p;  a2 += g * g;
        const float km = ((pt > 0.5f) && effb) ? 1.f : 0.f;
        const float pp = pk * km;
        const float gg = ((tk > 0.5f) ? 1.f : 0.f) * km;
        a3 += pp * gg;  a4 += pp * pp;  a5 += gg * gg;
    }
    red[0][tid] = a0; red[1][tid] = a1; red[2][tid] = a2;
    red[3][tid] = a3; red[4][tid] = a4; red[5][tid] = a5;
    __syncthreads();
    for (int off = NTHR / 2; off > 0; off >>= 1) {
        if (tid < off) {
            #pragma unroll
            for (int s = 0; s < 6; ++s) red[s][tid] += red[s][tid + off];
        }
        __syncthreads();
    }
    if (tid == 0) {
        #pragma unroll
        for (int s = 0; s < 6; ++s)
            partials[((size_t)n * 6 + s) * BPB + blockIdx.x] = red[s][0];
    }
}

// ---------------- final reduction via V_WMMA_F32_16X16X4_F32 ----------------
// A = 64 partials (16x4), B = ones(4x16) -> D[m][n] = rowsum(A,m).
// Sum of column 0 over all M = exact f32 sum of all 64 partials.
__global__ void tkl_finalize(const float* __restrict__ partials,
                             float* __restrict__ out) {
    const int n = blockIdx.x;
    const int l = threadIdx.x;                // 0..31, wave32, EXEC all ones
    const int i0 = (l & 15) * 4 + (l >> 4) * 2;  // bijection lanes/vgprs -> 0..63

    v2f B; B.x = 1.0f; B.y = 1.0f;
    float tot[6];
    #pragma unroll
    for (int s = 0; s < 6; ++s) {
        const float* p = partials + ((size_t)n * 6 + s) * BPB;
        v2f A; A.x = p[i0]; A.y = p[i0 + 1];
        v8f C = {};
        v8f D = __builtin_amdgcn_wmma_f32_16x16x4_f32(
            /*neg_a=*/false, A, /*neg_b=*/false, B,
            /*c_mod=*/(short)0, C, /*reuse_a=*/false, /*reuse_b=*/false);
        float srow = D[0] + D[1] + D[2] + D[3] + D[4] + D[5] + D[6] + D[7];
        // lane 0 holds M=0..7 @ N=0, lane 16 holds M=8..15 @ N=0
        tot[s] = __shfl(srow, 0, 32) + __shfl(srow, 16, 32);
    }
    if (l == 0) {
        out[n]           = 1.0f - (2.0f * tot[0] + SMOOTHF) / (tot[1] + tot[2] + SMOOTHF);
        out[N_BATCH + n] = 1.0f - (2.0f * tot[3] + SMOOTHF) / (tot[4] + tot[5] + SMOOTHF);
    }
}

extern "C" void kernel_launch(void* const* d_in, const int* in_sizes, int n_in,
                              void* d_out, int out_size, void* d_ws, size_t ws_size,
                              hipStream_t stream) {
    const float* preds   = (const float*)d_in[0];
    const float* targets = (const float*)d_in[1];
    const float* eff     = (const float*)d_in[2];

    // workspace layout (words)
    unsigned* hist   = (unsigned*)d_ws;            // 16*256
    unsigned* posn   = hist   + N_BATCH * 256;     // 16
    unsigned* negav  = posn   + N_BATCH;           // 16
    unsigned* prefix = negav  + N_BATCH;           // 16
    unsigned* krem   = prefix + N_BATCH;           // 16
    unsigned* deg    = krem   + N_BATCH;           // 16
    float*    thr    = (float*)(deg + N_BATCH);    // 16
    float*    part   = thr + N_BATCH;              // 16*6*64

    const int zero_words = N_BATCH * 256 + 2 * N_BATCH;  // hist + posn + negav
    tkl_zero_ws<<<(zero_words + 255) / 256, 256, 0, stream>>>(hist, zero_words);

    dim3 g(BPB, N_BATCH);
    tkl_hist0<<<g, NTHR, 0, stream>>>(preds, targets, eff, hist, posn, negav);
    tkl_select<<<N_BATCH, 256, 0, stream>>>(hist, posn, negav, prefix, krem, deg, thr, 0);
    tkl_hist<<<g, NTHR, 0, stream>>>(preds, targets, prefix, hist, 16, 0xFF000000u);
    tkl_select<<<N_BATCH, 256, 0, stream>>>(hist, posn, negav, prefix, krem, deg, thr, 1);
    tkl_hist<<<g, NTHR, 0, stream>>>(preds, targets, prefix, hist, 8, 0xFFFF0000u);
    tkl_select<<<N_BATCH, 256, 0, stream>>>(hist, posn, negav, prefix, krem, deg, thr, 2);
    tkl_hist<<<g, NTHR, 0, stream>>>(preds, targets, prefix, hist, 0, 0xFFFFFF00u);
    tkl_select<<<N_BATCH, 256, 0, stream>>>(hist, posn, negav, prefix, krem, deg, thr, 3);

    tkl_dice<<<g, NTHR, 0, stream>>>(preds, targets, eff, thr, deg, part);
    tkl_finalize<<<N_BATCH, 32, 0, stream>>>(part, (float*)d_out);
}